// MambaRefiner_88596585382359
// MI455X (gfx1250) — compile-verified
//
#include <hip/hip_runtime.h>
#include <hip/hip_bf16.h>

// ---------------- problem constants (from reference) ----------------
#define B_      2
#define C_      64
#define T_      512
#define CH      8
#define D_MODEL 512          // C_*CH
#define D_IN    1024         // 2*D_MODEL
#define D_ST    128
#define DT_R    32
#define KCONV   3
#define LAYERS  2
#define BT      (B_*T_)      // 1024 token rows
#define XPROJ_N (DT_R + 2*D_ST)   // 288

typedef float v2f __attribute__((ext_vector_type(2)));
typedef float v8f __attribute__((ext_vector_type(8)));
typedef unsigned int u32x4 __attribute__((ext_vector_type(4)));
typedef int i32x8 __attribute__((ext_vector_type(8)));
typedef int i32x4 __attribute__((ext_vector_type(4)));

__device__ __forceinline__ float fsigmoid(float x) { return 1.0f / (1.0f + __expf(-x)); }
__device__ __forceinline__ float fsilu(float x)    { return x * fsigmoid(x); }
__device__ __forceinline__ float fsoftplus(float x){ return (x > 20.0f) ? x : log1pf(__expf(x)); }

// LDS byte offset of a __shared__ object (generic -> addrspace(3) -> int)
__device__ __forceinline__ unsigned int lds_addr_of(const void* p) {
    return (unsigned int)(size_t)(__attribute__((address_space(3))) const void*)p;
}

// =====================================================================
// Tensor Data Mover: async 2D tile load Global -> LDS  (ISA ch.8 D#)
//   tile_cols elements per row (f32), tile_rows rows, row stride in
//   elements. Rows >= tensor_rows are hardware zero-filled (OOB reads).
//   pad33: insert 1 dword LDS pad every 32 dwords (33-dword rows).
// Tracked by TENSORcnt; caller waits with s_wait_tensorcnt.
// =====================================================================
__device__ __forceinline__ void tdm_load_tile(const float* gsrc, unsigned int lds_byte,
                                              unsigned int tile_cols, unsigned int tile_rows,
                                              unsigned int tensor_rows,
                                              unsigned long long row_stride_elems,
                                              bool pad33)
{
    unsigned long long ga = (unsigned long long)(size_t)gsrc;
    u32x4 g0;
    g0[0] = 1u;                                            // count = 1 (valid D#)
    g0[1] = lds_byte;                                      // lds_addr
    g0[2] = (unsigned int)ga;                              // global_addr[31:0]
    g0[3] = (unsigned int)((ga >> 32) & 0x1FFFFFFull)      // global_addr[56:32]
          | (2u << 30);                                    // type = 2 ("image")
    unsigned int w0 = (2u << 16);                          // data_size = 4 bytes
    if (pad33) w0 |= (1u << 20)                            // pad_enable
                   | (4u << 22);                           // pad_interval: 32 dwords
                                                           // pad_amount = 0 -> 1 dword
    i32x8 g1;
    g1[0] = (int)w0;
    g1[1] = (int)((tile_cols & 0xFFFFu) << 16);            // tensor_dim0[15:0]
    g1[2] = (int)(((tile_cols >> 16) & 0xFFFFu)            // tensor_dim0[31:16]
          | ((tensor_rows & 0xFFFFu) << 16));              // tensor_dim1[15:0]
    g1[3] = (int)(((tensor_rows >> 16) & 0xFFFFu)          // tensor_dim1[31:16]
          | ((tile_cols & 0xFFFFu) << 16));                // tile_dim0
    g1[4] = (int)(tile_rows & 0xFFFFu);                    // tile_dim1 (tile_dim2 = 0)
    g1[5] = (int)(unsigned int)(row_stride_elems & 0xFFFFFFFFull);   // dim0_stride lo
    g1[6] = (int)(unsigned int)((row_stride_elems >> 32) & 0xFFFFull);
    g1[7] = 0;
    i32x4 z4 = {0, 0, 0, 0};                               // group2 / group3 (2-D: unused)
    i32x8 z8 = {0, 0, 0, 0, 0, 0, 0, 0};                   // trailing operand (6-arg form)
    __builtin_amdgcn_tensor_load_to_lds(g0, g1, z4, z4, z8, 0);
}

// =====================================================================
// WMMA fp32 GEMM:  Cout[M,N] = act( A[M,K] @ W[N,K]^T + bias )
// TDM double-buffered LDS staging; v_wmma_f32_16x16x4_f32 compute.
// mode bit0: +bias[n]; bit1: softplus; bit2: permuted (B,C,T,CH) store.
// Block 256 = 8 waves; tile 32(M) x 64(N); K slab 32; K % 32 == 0.
// =====================================================================
__global__ __launch_bounds__(256)
void wmma_gemm_kernel(const float* __restrict__ A, int lda,
                      const float* __restrict__ W, int ldw,
                      const float* __restrict__ bias,
                      float* __restrict__ Cout, int ldc,
                      int M, int N, int K, int mode)
{
    __shared__ float sA[2][32 * 33];   // TDM pads rows to 33 dwords
    __shared__ float sW[2][64 * 33];

    const int tid   = threadIdx.x;
    const int lane  = tid & 31;
    const int wv    = tid >> 5;        // wave 0..7
    const int mr    = wv >> 2;         // 0..1 : 16-row sub-tile
    const int nc    = wv & 3;          // 0..3 : 16-col sub-tile
    const int mblk  = blockIdx.y * 32;
    const int nblk  = blockIdx.x * 64;
    const int lhalf = lane >> 4;       // K-half selector (16x4 f32 A/B layout)
    const int lmod  = lane & 15;       // M (or N) index within fragment

    const int nslab = K >> 5;

    if (wv == 0) {                     // TDM issue: one wave, EXEC-independent DMA
        tdm_load_tile(&A[(size_t)mblk * lda], lds_addr_of(&sA[0][0]),
                      32, 32, (unsigned)(M - mblk), (unsigned long long)lda, true);
        tdm_load_tile(&W[(size_t)nblk * ldw], lds_addr_of(&sW[0][0]),
                      32, 64, (unsigned)(N - nblk), (unsigned long long)ldw, true);
    }

    v8f acc = {};

    for (int s = 0; s < nslab; ++s) {
        const int cur = s & 1;
        if (wv == 0) {
            if (s + 1 < nslab) {       // prefetch next slab into other buffer
                const int kb = (s + 1) << 5;
                tdm_load_tile(&A[(size_t)mblk * lda + kb], lds_addr_of(&sA[cur ^ 1][0]),
                              32, 32, (unsigned)(M - mblk), (unsigned long long)lda, true);
                tdm_load_tile(&W[(size_t)nblk * ldw + kb], lds_addr_of(&sW[cur ^ 1][0]),
                              32, 64, (unsigned)(N - nblk), (unsigned long long)ldw, true);
                __builtin_amdgcn_s_wait_tensorcnt(2);   // current slab's 2 loads done
            } else {
                __builtin_amdgcn_s_wait_tensorcnt(0);
            }
        }
        __syncthreads();               // LDS tiles visible to all 8 waves

        #pragma unroll
        for (int kk = 0; kk < 32; kk += 4) {
            v2f a, b;
            a.x = sA[cur][(mr * 16 + lmod) * 33 + kk + lhalf * 2 + 0];
            a.y = sA[cur][(mr * 16 + lmod) * 33 + kk + lhalf * 2 + 1];
            b.x = sW[cur][(nc * 16 + lmod) * 33 + kk + lhalf * 2 + 0];
            b.y = sW[cur][(nc * 16 + lmod) * 33 + kk + lhalf * 2 + 1];
            acc = __builtin_amdgcn_wmma_f32_16x16x4_f32(
                      false, a, false, b, (short)0, acc, false, false);
        }
        __syncthreads();               // done reading before next overwrite
    }

    // ---- store 16x16 tile: D layout M = r + 8*lhalf, N = lmod ----
    const int n = nblk + nc * 16 + lmod;
    if (n < N) {
        const float bval = (mode & 1) ? bias[n] : 0.0f;
        #pragma unroll
        for (int r = 0; r < 8; ++r) {
            int m   = mblk + mr * 16 + r + 8 * lhalf;
            float v = acc[r] + bval;
            if (mode & 2) v = fsoftplus(v);
            if (mode & 4) {
                // out.reshape(B,T,C,CH).transpose(0,2,1,3) -> (B,C,T,CH)
                int b_ = m >> 9, t = m & (T_ - 1);
                int cc = n >> 3, ch = n & 7;
                Cout[(((size_t)(b_ * C_ + cc)) * T_ + t) * CH + ch] = v;
            } else {
                Cout[(size_t)m * ldc + n] = v;
            }
        }
    }
}

// =====================================================================
// z_q (B,C,T,CH) -> u (B,T,D_MODEL)
// =====================================================================
__global__ __launch_bounds__(256)
void permute_in_kernel(const float* __restrict__ zq, float* __restrict__ u)
{
    int idx = blockIdx.x * 256 + threadIdx.x;          // over BT*D_MODEL
    if (idx >= BT * D_MODEL) return;
    int m = idx >> 9;
    int n = idx & (D_MODEL - 1);
    int b = m >> 9, t = m & (T_ - 1);
    u[idx] = zq[(((size_t)(b * C_ + (n >> 3))) * T_ + t) * CH + (n & 7)];
}

// =====================================================================
// Causal depthwise conv (K=3) on x = xz[:, :D_IN], + bias, SiLU
// =====================================================================
__global__ __launch_bounds__(256)
void conv_silu_kernel(const float* __restrict__ xz,   // (BT, 2*D_IN)
                      const float* __restrict__ cw,   // (D_IN, KCONV)
                      const float* __restrict__ cb,   // (D_IN)
                      float* __restrict__ xq)         // (BT, D_IN)
{
    int idx = blockIdx.x * 256 + threadIdx.x;
    if (idx >= BT * D_IN) return;
    int m = idx >> 10;
    int d = idx & (D_IN - 1);
    int t = m & (T_ - 1);
    float acc = cb[d];
    #pragma unroll
    for (int k = 0; k < KCONV; ++k) {
        int ts = t + k - (KCONV - 1);
        if (ts >= 0)
            acc += xz[(size_t)(m + k - (KCONV - 1)) * (2 * D_IN) + d] * cw[d * KCONV + k];
    }
    xq[idx] = fsilu(acc);
}

// =====================================================================
// Selective scan. One wave32 per (b,d); lane owns 4 of 128 states.
// B_t/C_t chunk (32 t-steps x 256 f32, stride 288) fetched by TDM into
// double-buffered LDS, shared by the 8 waves; next chunk prefetched
// during compute. y includes the D-skip.
// =====================================================================
__global__ __launch_bounds__(256)
void scan_kernel(const float* __restrict__ xq,     // (BT, D_IN)
                 const float* __restrict__ xdbl,   // (BT, 288): dt|B|C
                 const float* __restrict__ delta,  // (BT, D_IN) softplus'ed
                 const float* __restrict__ alog,   // (D_IN, D_ST)
                 const float* __restrict__ Dskip,  // (D_IN)
                 float* __restrict__ y)            // (BT, D_IN)
{
    __shared__ float sBC[2][32 * 2 * D_ST];        // [t][0:128]=B, [128:256]=C

    const int tid  = threadIdx.x;
    const int lane = tid & 31;
    const int wv   = tid >> 5;
    const int b    = blockIdx.y;
    const int d    = blockIdx.x * 8 + wv;
    const int s0   = lane * 4;

    const float A0 = -__expf(alog[(size_t)d * D_ST + s0 + 0]);
    const float A1 = -__expf(alog[(size_t)d * D_ST + s0 + 1]);
    const float A2 = -__expf(alog[(size_t)d * D_ST + s0 + 2]);
    const float A3 = -__expf(alog[(size_t)d * D_ST + s0 + 3]);
    const float dsk = Dskip[d];

    if (wv == 0) {
        tdm_load_tile(&xdbl[(size_t)(b * T_) * XPROJ_N + DT_R],
                      lds_addr_of(&sBC[0][0]), 2 * D_ST, 32,
                      (unsigned)(BT - b * T_), XPROJ_N, false);
    }

    float h0 = 0.f, h1 = 0.f, h2 = 0.f, h3 = 0.f;
    const int nchunk = T_ / 32;

    for (int ci = 0; ci < nchunk; ++ci) {
        const int cur = ci & 1;
        const int t0  = ci * 32;
        if (wv == 0) {
            if (ci + 1 < nchunk) {
                tdm_load_tile(&xdbl[(size_t)(b * T_ + t0 + 32) * XPROJ_N + DT_R],
                              lds_addr_of(&sBC[cur ^ 1][0]), 2 * D_ST, 32,
                              (unsigned)(BT - (b * T_ + t0 + 32)), XPROJ_N, false);
                __builtin_amdgcn_s_wait_tensorcnt(1);
            } else {
                __builtin_amdgcn_s_wait_tensorcnt(0);
            }
        }
        __syncthreads();

        for (int tt = 0; tt < 32; ++tt) {
            const int m = b * T_ + t0 + tt;
            const float dv = delta[(size_t)m * D_IN + d];   // wave-uniform
            const float xv = xq[(size_t)m * D_IN + d];
            const float dx = dv * xv;
            const float* bc = &sBC[cur][tt * 2 * D_ST];
            h0 = __expf(dv * A0) * h0 + dx * bc[s0 + 0];
            h1 = __expf(dv * A1) * h1 + dx * bc[s0 + 1];
            h2 = __expf(dv * A2) * h2 + dx * bc[s0 + 2];
            h3 = __expf(dv * A3) * h3 + dx * bc[s0 + 3];
            float p = h0 * bc[D_ST + s0 + 0]
                    + h1 * bc[D_ST + s0 + 1]
                    + h2 * bc[D_ST + s0 + 2]
                    + h3 * bc[D_ST + s0 + 3];
            #pragma unroll
            for (int off = 16; off > 0; off >>= 1)
                p += __shfl_xor(p, off, 32);
            if (lane == 0)
                y[(size_t)m * D_IN + d] = p + xv * dsk;
        }
        __syncthreads();
    }
}

// =====================================================================
// yg = y * silu(z),  z = xz[:, D_IN:2*D_IN]
// =====================================================================
__global__ __launch_bounds__(256)
void gate_kernel(const float* __restrict__ y, const float* __restrict__ xz,
                 float* __restrict__ yg)
{
    int idx = blockIdx.x * 256 + threadIdx.x;
    if (idx >= BT * D_IN) return;
    int m = idx >> 10;
    int d = idx & (D_IN - 1);
    yg[idx] = y[idx] * fsilu(xz[(size_t)m * (2 * D_IN) + D_IN + d]);
}

// =====================================================================
extern "C" void kernel_launch(void* const* d_in, const int* in_sizes, int n_in,
                              void* d_out, int out_size, void* d_ws, size_t ws_size,
                              hipStream_t stream)
{
    const float* z_q   = (const float*)d_in[0];
    const float* ipw   = (const float*)d_in[1];   // (L, 2*D_IN, D_MODEL)
    const float* cw    = (const float*)d_in[2];   // (L, D_IN, K)
    const float* cb    = (const float*)d_in[3];   // (L, D_IN)
    const float* xpw   = (const float*)d_in[4];   // (L, 288, D_IN)
    const float* dpw   = (const float*)d_in[5];   // (L, D_IN, DT_R)
    const float* dpb   = (const float*)d_in[6];   // (L, D_IN)
    const float* alog  = (const float*)d_in[7];   // (L, D_IN, D_ST)
    const float* Dsk   = (const float*)d_in[8];   // (L, D_IN)
    const float* opw   = (const float*)d_in[9];   // (L, D_MODEL, D_IN)
    const float* fcw   = (const float*)d_in[10];  // (D_MODEL, D_MODEL)
    const float* fcb   = (const float*)d_in[11];  // (D_MODEL)
    float* out = (float*)d_out;

    // workspace carve-up (floats); total ~7.11M floats = ~28.4 MB
    float* u    = (float*)d_ws;                  // (BT, D_MODEL)
    float* xz   = u    + (size_t)BT * D_MODEL;   // (BT, 2*D_IN)
    float* xq   = xz   + (size_t)BT * 2 * D_IN;  // (BT, D_IN)
    float* xdbl = xq   + (size_t)BT * D_IN;      // (BT, 288)
    float* dlt  = xdbl + (size_t)BT * XPROJ_N;   // (BT, D_IN)
    float* yy   = dlt  + (size_t)BT * D_IN;      // (BT, D_IN)
    float* yg   = yy   + (size_t)BT * D_IN;      // (BT, D_IN)

    dim3 blk(256);

    permute_in_kernel<<<(BT * D_MODEL + 255) / 256, blk, 0, stream>>>(z_q, u);

    for (int l = 0; l < LAYERS; ++l) {
        const float* ipw_l  = ipw  + (size_t)l * (2 * D_IN) * D_MODEL;
        const float* cw_l   = cw   + (size_t)l * D_IN * KCONV;
        const float* cb_l   = cb   + (size_t)l * D_IN;
        const float* xpw_l  = xpw  + (size_t)l * XPROJ_N * D_IN;
        const float* dpw_l  = dpw  + (size_t)l * D_IN * DT_R;
        const float* dpb_l  = dpb  + (size_t)l * D_IN;
        const float* alog_l = alog + (size_t)l * D_IN * D_ST;
        const float* Dsk_l  = Dsk  + (size_t)l * D_IN;
        const float* opw_l  = opw  + (size_t)l * D_MODEL * D_IN;

        // xz = u @ ipw^T          (1024 x 2048, K=512)
        wmma_gemm_kernel<<<dim3(2 * D_IN / 64, BT / 32), blk, 0, stream>>>(
            u, D_MODEL, ipw_l, D_MODEL, nullptr, xz, 2 * D_IN,
            BT, 2 * D_IN, D_MODEL, 0);

        // x = silu(causal_dwconv(xz[:, :D_IN]) + cb)
        conv_silu_kernel<<<(BT * D_IN + 255) / 256, blk, 0, stream>>>(xz, cw_l, cb_l, xq);

        // x_dbl = x @ xpw^T       (1024 x 288, K=1024)
        wmma_gemm_kernel<<<dim3((XPROJ_N + 63) / 64, BT / 32), blk, 0, stream>>>(
            xq, D_IN, xpw_l, D_IN, nullptr, xdbl, XPROJ_N,
            BT, XPROJ_N, D_IN, 0);

        // delta = softplus(dt @ dpw^T + dpb)   (1024 x 1024, K=32, lda=288)
        wmma_gemm_kernel<<<dim3(D_IN / 64, BT / 32), blk, 0, stream>>>(
            xdbl, XPROJ_N, dpw_l, DT_R, dpb_l, dlt, D_IN,
            BT, D_IN, DT_R, 1 | 2);

        // selective scan (+ D skip)
        scan_kernel<<<dim3(D_IN / 8, B_), blk, 0, stream>>>(
            xq, xdbl, dlt, alog_l, Dsk_l, yy);

        // gate: yg = y * silu(z)
        gate_kernel<<<(BT * D_IN + 255) / 256, blk, 0, stream>>>(yy, xz, yg);

        // u = yg @ opw^T          (1024 x 512, K=1024)  — next layer input
        wmma_gemm_kernel<<<dim3(D_MODEL / 64, BT / 32), blk, 0, stream>>>(
            yg, D_IN, opw_l, D_IN, nullptr, u, D_MODEL,
            BT, D_MODEL, D_IN, 0);
    }

    // out = u @ fcw^T + fcb, stored directly in (B,C,T,CH) layout
    wmma_gemm_kernel<<<dim3(D_MODEL / 64, BT / 32), blk, 0, stream>>>(
        u, D_MODEL, fcw, D_MODEL, fcb, out, D_MODEL,
        BT, D_MODEL, D_MODEL, 1 | 4);
}